// RotaryPositionalEmbedding_2413771620710
// MI455X (gfx1250) — compile-verified
//
#include <hip/hip_runtime.h>

// RoPE + sequence gather for (B=4, H=16, S=8192, D=64) fp32 on MI455X (gfx1250).
//
// Roofline: ~270 MB HBM traffic vs ~100 MFLOP -> bandwidth bound, floor ~11.6 us
// at 23.3 TB/s. The gather creates a 2-deep dependent load chain (pos -> x), so
// the binding constraints are memory-level parallelism per wave and issue
// efficiency, not ALU throughput. WMMA cannot apply at 0.37 flop/byte.
//
// Structure: 4 consecutive rows per 16-lane group.
//  * One aligned b128 load fetches all 4 token positions (round trip #1
//    amortized 4x; the 4 slots never cross a batch*head boundary since S is a
//    power of two divisible by 4).
//  * All 12 data loads (4x b128 x, 8x b64 cos/sin) independent, issued before
//    any wait -> ~4 KB of reads in flight per wave.
//  * All addresses are 32-bit byte offsets off SGPR bases (saddr + voffset
//    form, like the compiler already picked for the pos load) -> fewer VALU
//    ops and VGPRs than 64-bit address pairs -> better occupancy for latency
//    hiding.
//  * Non-temporal b128 stores: output is write-once; keep the 192 MB L2 for
//    gathered x rows (~37% re-read) and the 2 MB cos/sin tables.

#define D_K    64
#define HALF_D 32

typedef float v4f __attribute__((ext_vector_type(4)));
typedef float v2f __attribute__((ext_vector_type(2)));
typedef int   v4i __attribute__((ext_vector_type(4)));

__device__ __forceinline__ v4f ld128(const float* base, unsigned int byteOff) {
    return *reinterpret_cast<const v4f*>(reinterpret_cast<const char*>(base) + byteOff);
}
__device__ __forceinline__ v2f ld64(const float* base, unsigned int byteOff) {
    return *reinterpret_cast<const v2f*>(reinterpret_cast<const char*>(base) + byteOff);
}
__device__ __forceinline__ void st128_nt(float* base, unsigned int byteOff, v4f v) {
    __builtin_nontemporal_store(
        v, reinterpret_cast<v4f*>(reinterpret_cast<char*>(base) + byteOff));
}

__global__ __launch_bounds__(256) void rope_gather_u4(
    const float* __restrict__ x,      // (BH, S, 64)
    const float* __restrict__ cosT,   // (MAX_SEQ_LEN, 32)
    const float* __restrict__ sinT,   // (MAX_SEQ_LEN, 32)
    const int*   __restrict__ pos,    // (S,)
    float*       __restrict__ out,    // (BH, S, 64)
    unsigned int sShift,              // log2(S)
    unsigned int sMask,               // S - 1
    unsigned int nGroups)             // (B*H*S) / 4 row groups
{
    unsigned int tid    = blockIdx.x * blockDim.x + threadIdx.x;
    unsigned int g      = tid >> 4;        // group of 4 consecutive rows
    unsigned int lane16 = tid & 15u;       // which float4 within each row
    if (g >= nGroups) return;

    unsigned int r0 = g * 4u;              // first output row of the group
    unsigned int i0 = r0 & sMask;          // 4-aligned; group stays inside one bh
    unsigned int bh = r0 >> sShift;

    // Round trip #1: one aligned 128-bit load -> 4 gathered positions.
    const v4i pv = *reinterpret_cast<const v4i*>(pos + i0);
    unsigned int p0 = (unsigned int)pv.x, p1 = (unsigned int)pv.y;
    unsigned int p2 = (unsigned int)pv.z, p3 = (unsigned int)pv.w;

    // 32-bit BYTE offsets (max 134 MB < 2^31) -> saddr + voffset addressing.
    const unsigned int xbase = (bh << (sShift + 8u)) + (lane16 << 4u); // row 256B, lane 16B
    const unsigned int tlane = lane16 << 3u;                           // lane 8B in tables

    const unsigned int xo0 = xbase + (p0 << 8u);
    const unsigned int xo1 = xbase + (p1 << 8u);
    const unsigned int xo2 = xbase + (p2 << 8u);
    const unsigned int xo3 = xbase + (p3 << 8u);
    const unsigned int to0 = (p0 << 7u) + tlane;   // shared by cos & sin
    const unsigned int to1 = (p1 << 7u) + tlane;
    const unsigned int to2 = (p2 << 7u) + tlane;
    const unsigned int to3 = (p3 << 7u) + tlane;

    // Round trip #2: 12 independent loads, all issued before any use.
    v4f xv0 = ld128(x, xo0);
    v4f xv1 = ld128(x, xo1);
    v4f xv2 = ld128(x, xo2);
    v4f xv3 = ld128(x, xo3);
    v2f cv0 = ld64(cosT, to0);
    v2f cv1 = ld64(cosT, to1);
    v2f cv2 = ld64(cosT, to2);
    v2f cv3 = ld64(cosT, to3);
    v2f sv0 = ld64(sinT, to0);
    v2f sv1 = ld64(sinT, to1);
    v2f sv2 = ld64(sinT, to2);
    v2f sv3 = ld64(sinT, to3);

    const unsigned int ob = (r0 << 8u) + (lane16 << 4u);

    // Pair j: oe = xe*c - xo*s ; oo = xe*s + xo*c   (v_fma_f32)
    v4f o0, o1, o2, o3;
    o0.x = fmaf(xv0.x, cv0.x, -(xv0.y * sv0.x));
    o0.y = fmaf(xv0.x, sv0.x,   xv0.y * cv0.x);
    o0.z = fmaf(xv0.z, cv0.y, -(xv0.w * sv0.y));
    o0.w = fmaf(xv0.z, sv0.y,   xv0.w * cv0.y);
    st128_nt(out, ob + 0u * 256u, o0);

    o1.x = fmaf(xv1.x, cv1.x, -(xv1.y * sv1.x));
    o1.y = fmaf(xv1.x, sv1.x,   xv1.y * cv1.x);
    o1.z = fmaf(xv1.z, cv1.y, -(xv1.w * sv1.y));
    o1.w = fmaf(xv1.z, sv1.y,   xv1.w * cv1.y);
    st128_nt(out, ob + 1u * 256u, o1);

    o2.x = fmaf(xv2.x, cv2.x, -(xv2.y * sv2.x));
    o2.y = fmaf(xv2.x, sv2.x,   xv2.y * cv2.x);
    o2.z = fmaf(xv2.z, cv2.y, -(xv2.w * sv2.y));
    o2.w = fmaf(xv2.z, sv2.y,   xv2.w * cv2.y);
    st128_nt(out, ob + 2u * 256u, o2);

    o3.x = fmaf(xv3.x, cv3.x, -(xv3.y * sv3.x));
    o3.y = fmaf(xv3.x, sv3.x,   xv3.y * cv3.x);
    o3.z = fmaf(xv3.z, cv3.y, -(xv3.w * sv3.y));
    o3.w = fmaf(xv3.z, sv3.y,   xv3.w * cv3.y);
    st128_nt(out, ob + 3u * 256u, o3);
}

// Generic fallback (only used if S is not a power of two or nRows % 4 != 0).
__global__ __launch_bounds__(256) void rope_gather_generic(
    const float* __restrict__ x,
    const float* __restrict__ cosT,
    const float* __restrict__ sinT,
    const int*   __restrict__ pos,
    float*       __restrict__ out,
    unsigned int S,
    unsigned int nRows)
{
    unsigned int tid    = blockIdx.x * blockDim.x + threadIdx.x;
    unsigned int r      = tid >> 4;
    unsigned int lane16 = tid & 15u;
    if (r >= nRows) return;

    unsigned int i  = r % S;
    unsigned int bh = r / S;
    unsigned int p  = (unsigned int)pos[i];

    unsigned int xoff = ((bh * S + p) * D_K + lane16 * 4u) * 4u;
    unsigned int toff = (p * HALF_D + lane16 * 2u) * 4u;

    const v4f xv = ld128(x, xoff);
    const v2f cv = ld64(cosT, toff);
    const v2f sv = ld64(sinT, toff);

    v4f o;
    o.x = fmaf(xv.x, cv.x, -(xv.y * sv.x));
    o.y = fmaf(xv.x, sv.x,   xv.y * cv.x);
    o.z = fmaf(xv.z, cv.y, -(xv.w * sv.y));
    o.w = fmaf(xv.z, sv.y,   xv.w * cv.y);

    st128_nt(out, (r * D_K + lane16 * 4u) * 4u, o);
}

extern "C" void kernel_launch(void* const* d_in, const int* in_sizes, int n_in,
                              void* d_out, int out_size, void* d_ws, size_t ws_size,
                              hipStream_t stream)
{
    (void)n_in; (void)d_ws; (void)ws_size; (void)out_size;

    const float* x    = (const float*)d_in[0];   // (B,H,S,64) fp32
    const float* cosT = (const float*)d_in[1];   // (MAX_SEQ_LEN,32) fp32
    const float* sinT = (const float*)d_in[2];   // (MAX_SEQ_LEN,32) fp32
    const int*   pos  = (const int*)  d_in[3];   // (S,) int32
    float*       out  = (float*)d_out;

    const unsigned int S     = (unsigned int)in_sizes[3];          // 8192
    const unsigned int nRows = (unsigned int)(in_sizes[0] / D_K);  // B*H*S = 524288

    const unsigned int threads = 256;                              // 8 wave32 / block

    if ((S & (S - 1u)) == 0u && S >= 4u && (nRows & 3u) == 0u) {
        unsigned int sh = 0;
        while ((1u << sh) < S) ++sh;
        const unsigned int nGroups = nRows / 4u;                   // 131072
        const unsigned int total   = nGroups * 16u;                // 16 lanes per group
        const unsigned int blocks  = (total + threads - 1) / threads;
        rope_gather_u4<<<blocks, threads, 0, stream>>>(
            x, cosT, sinT, pos, out, sh, S - 1u, nGroups);
    } else {
        const unsigned int total  = nRows * 16u;
        const unsigned int blocks = (total + threads - 1) / threads;
        rope_gather_generic<<<blocks, threads, 0, stream>>>(
            x, cosT, sinT, pos, out, S, nRows);
    }
}